// DensityDenoiser_67405216743690
// MI455X (gfx1250) — compile-verified
//
#include <hip/hip_runtime.h>
#include <math.h>

// Shapes fixed by the reference: H=8, DK=16, C=8, M=4.
#define H    8
#define DK   16
#define CM   32          // C*M
#define HCM  256         // H*C*M
#define SCALE 0.08838834764831845f   // 1/sqrt(H*DK) = 1/sqrt(128)

typedef float v2f __attribute__((ext_vector_type(2)));
typedef float v8f __attribute__((ext_vector_type(8)));

// Order-preserving float <-> uint mapping so unsigned atomicMax == float max
// (works for negatives and -inf).
__device__ __forceinline__ unsigned map_f2u(float f) {
  unsigned u = __float_as_uint(f);
  return (u & 0x80000000u) ? ~u : (u | 0x80000000u);
}
__device__ __forceinline__ float unmap_u2f(unsigned m) {
  unsigned u = (m & 0x80000000u) ? (m & 0x7FFFFFFFu) : ~m;
  return __uint_as_float(u);
}

// ---------------------------------------------------------------------------
// Pass 0: zero the accumulation targets (out region of d_out, segmax/denom ws)
// ---------------------------------------------------------------------------
__global__ void k_init(float* __restrict__ out, unsigned* __restrict__ segmax,
                       float* __restrict__ denom, size_t nOut, size_t nSeg) {
  size_t i = (size_t)blockIdx.x * blockDim.x + threadIdx.x;
  size_t stride = (size_t)gridDim.x * blockDim.x;
  for (size_t k = i; k < nOut; k += stride) out[k] = 0.0f;
  for (size_t k = i; k < nSeg; k += stride) { segmax[k] = 0u; denom[k] = 0.0f; }
}

// ---------------------------------------------------------------------------
// Pass 1: logits[r] = scale * dot(key[r,:], query[dst[r/H], r%H, :])
// One wave handles 16 flattened (e,h) rows via diag(K * Q^T) using
// 4x V_WMMA_F32_16X16X4_F32 accumulated over K=16.
//
// A (16x4 f32) layout: lanes 0-15 = rows, VGPR0/1 = K 0,1; lanes 16-31 = K 2,3.
// B fed as Q^T in the mirrored layout (lane = column / q-row).
// D (16x16 f32) layout: VGPR v holds (M=v, N=lane) for lanes 0-15 and
// (M=v+8, N=lane-16) for lanes 16-31 -> diagonal row m lives in lane m
// (acc[m], m<8) and lane m+16 (acc[m-8], m>=8).
// ---------------------------------------------------------------------------
__global__ void k_logits(const float* __restrict__ key,
                         const float* __restrict__ query,
                         const int*   __restrict__ dst,
                         float*       __restrict__ prelogits,
                         unsigned*    __restrict__ segmax,
                         int total) {
  const int lane = threadIdx.x & 31;
  const int wave = threadIdx.x >> 5;
  const int tile = blockIdx.x * (blockDim.x >> 5) + wave;
  const int base = tile << 4;
  if (base >= total) return;              // wave-uniform

  if (base + 16 <= total) {               // wave-uniform fast path, EXEC all 1s
    const int j  = lane & 15;             // tile row this lane services
    const int r  = base + j;              // flattened (e,h)
    const int e  = r >> 3;
    const int hh = r & 7;
    const int dd = __ldg(dst + e);
    const float* krow = key   + (size_t)r * DK;
    const float* qrow = query + ((size_t)dd * H + hh) * DK;
    const int koff = (lane >> 4) << 1;    // lanes 0-15: K+0,1 ; 16-31: K+2,3

    v8f acc = {};
#pragma unroll
    for (int c = 0; c < 4; ++c) {
      v2f a = *(const v2f*)(krow + c * 4 + koff);
      v2f b = *(const v2f*)(qrow + c * 4 + koff);
      acc = __builtin_amdgcn_wmma_f32_16x16x4_f32(false, a, false, b,
                                                  (short)0, acc, false, false);
    }

    // Extract D[m][m]: sel = lane & 7 picks the accumulator register.
    const int sel = lane & 7;
    float dv = acc[0];
    dv = (sel == 1) ? acc[1] : dv;
    dv = (sel == 2) ? acc[2] : dv;
    dv = (sel == 3) ? acc[3] : dv;
    dv = (sel == 4) ? acc[4] : dv;
    dv = (sel == 5) ? acc[5] : dv;
    dv = (sel == 6) ? acc[6] : dv;
    dv = (sel == 7) ? acc[7] : dv;

    if ((lane < 8) || (lane >= 24)) {     // lanes holding diagonal entries
      float lg = dv * SCALE;
      prelogits[r] = lg;
      atomicMax(&segmax[dd * H + hh], map_f2u(lg));
    }
  } else {
    // Scalar tail (never hit with E*H % 16 == 0, kept for generality).
    if (lane < 16) {
      int r = base + lane;
      if (r < total) {
        int e = r >> 3, hh = r & 7;
        int dd = __ldg(dst + e);
        const float* krow = key   + (size_t)r * DK;
        const float* qrow = query + ((size_t)dd * H + hh) * DK;
        float acc = 0.0f;
#pragma unroll
        for (int d = 0; d < DK; ++d) acc = fmaf(krow[d], qrow[d], acc);
        float lg = acc * SCALE;
        prelogits[r] = lg;
        atomicMax(&segmax[dd * H + hh], map_f2u(lg));
      }
    }
  }
}

// ---------------------------------------------------------------------------
// Pass 2: denom[dst,h] += exp(logit - segmax[dst,h])
// ---------------------------------------------------------------------------
__global__ void k_denom(const float* __restrict__ prelogits,
                        const int*   __restrict__ dst,
                        const unsigned* __restrict__ segmax,
                        float*       __restrict__ denom,
                        int total) {
  int r = blockIdx.x * blockDim.x + threadIdx.x;
  if (r >= total) return;
  int e = r >> 3, hh = r & 7;
  int dd = __ldg(dst + e);
  int s  = dd * H + hh;
  float ex = __expf(prelogits[r] - unmap_u2f(segmax[s]));
  unsafeAtomicAdd(&denom[s], ex);
}

// ---------------------------------------------------------------------------
// Pass 3: out[dst,h,c,m] += w * value[e,h,c,m], w = exp(l - max)/(denom+eps)
// value_e streamed once (819 MB); atomic targets live in L2 (51 MB).
// ---------------------------------------------------------------------------
__global__ void k_scatter(const float* __restrict__ value,
                          const float* __restrict__ prelogits,
                          const int*   __restrict__ dst,
                          const unsigned* __restrict__ segmax,
                          const float* __restrict__ denom,
                          float*       __restrict__ out,
                          int total) {
  int r = blockIdx.x * blockDim.x + threadIdx.x;
  if (r >= total) return;
  int e = r >> 3, hh = r & 7;
  int dd = __ldg(dst + e);
  int s  = dd * H + hh;
  float mx = unmap_u2f(segmax[s]);
  float w  = __expf(prelogits[r] - mx) / (denom[s] + 1e-9f);

  const float* v = value + (size_t)r * CM;
  // Stream-ahead hint (speculative prefetch is silently dropped if OOB).
  __builtin_prefetch(v + (size_t)CM * 4096, 0, 0);
  float* o = out + ((size_t)dd * H + hh) * CM;
#pragma unroll
  for (int i = 0; i < CM; i += 4) {
    float4 vv = *(const float4*)(v + i);
    unsafeAtomicAdd(o + i + 0, w * vv.x);
    unsafeAtomicAdd(o + i + 1, w * vv.y);
    unsafeAtomicAdd(o + i + 2, w * vv.z);
    unsafeAtomicAdd(o + i + 3, w * vv.w);
  }
}

// ---------------------------------------------------------------------------
extern "C" void kernel_launch(void* const* d_in, const int* in_sizes, int n_in,
                              void* d_out, int out_size, void* d_ws, size_t ws_size,
                              hipStream_t stream) {
  const float* key   = (const float*)d_in[0];   // [E,H,DK]
  const float* query = (const float*)d_in[1];   // [N,H,DK]
  const float* value = (const float*)d_in[2];   // [E,H,C,M]
  const int*   dst   = (const int*)d_in[3];     // [E]

  const int E  = in_sizes[3];
  const int Nn = in_sizes[1] / (H * DK);
  const int total = E * H;                      // flattened (e,h) count

  float*    out       = (float*)d_out;                  // [N, H*C, M]
  float*    prelogits = out + (size_t)Nn * HCM;         // [E, H]
  unsigned* segmax    = (unsigned*)d_ws;                // [N*H]
  float*    denom     = (float*)(segmax + (size_t)Nn * H); // [N*H]

  k_init<<<4096, 256, 0, stream>>>(out, segmax, denom,
                                   (size_t)Nn * HCM, (size_t)Nn * H);

  const int tiles  = (total + 15) / 16;
  const int blocks = (tiles + 7) / 8;           // 8 waves (tiles) per 256-thd block
  k_logits<<<blocks, 256, 0, stream>>>(key, query, dst, prelogits, segmax, total);

  const int b2 = (total + 255) / 256;
  k_denom<<<b2, 256, 0, stream>>>(prelogits, dst, segmax, denom, total);
  k_scatter<<<b2, 256, 0, stream>>>(value, prelogits, dst, segmax, denom, out, total);
}